// ScatterVerbsToHois_600_18408229831252
// MI455X (gfx1250) — compile-verified
//
#include <hip/hip_runtime.h>
#include <hip/hip_bf16.h>

// out[b, h] = verb_scores[b, hoi_to_verb[h]]
//   verb_scores: (BATCH, 117) f32
//   hoi_to_verb: (600,) int
//   out:         (BATCH, 600) f32
//
// Pure data movement: ~61 MB read + ~315 MB NT-stream write -> ~16 us floor
// at 23.3 TB/s. Strategy: async-stage the index map AND a 32-row input tile
// into LDS with gfx1250 GLOBAL_LOAD_ASYNC_TO_LDS_B128 (single s_wait_asynccnt,
// no per-element loadcnt stalls), gather from LDS, emit coalesced nontemporal
// float4 stores that stream past the caches (input stays L2-resident).

typedef __attribute__((ext_vector_type(4))) float v4f;

#define NV      117   // verbs per row
#define NH      600   // hois per row
#define TB      32    // batch rows per block
#define THREADS 256   // 8 wave32s

__global__ __launch_bounds__(THREADS)
void ScatterVerbsToHois_kernel(const float* __restrict__ verb,
                               const int*   __restrict__ hoi2verb,
                               float*       __restrict__ out,
                               int batch)
{
    __shared__ int   s_idx[NH];          // 2.4 KB
    __shared__ float s_rows[TB * NV];    // 14.6 KB, contiguous (same layout as global)

    const int tid  = threadIdx.x;
    const int row0 = blockIdx.x * TB;
    if (row0 >= batch) return;
    const int rows = min(TB, batch - row0);

    // ---- async-stage index map into LDS: 600 ints = 150 x B128 ----
    {
        const int nchunks = NH / 4;                  // 150
        for (int i = tid; i < nchunks; i += THREADS) {
            unsigned lds_off = (unsigned)(size_t)(&s_idx[i * 4]);
            unsigned goff    = (unsigned)(i * 16);
            asm volatile("global_load_async_to_lds_b128 %0, %1, %2"
                         :: "v"(lds_off), "v"(goff), "s"(hoi2verb) : "memory");
        }
    }

    // ---- async-stage the 32-row input tile into LDS ----
    const float* gbase = verb + (size_t)row0 * NV;   // tile base -> SGPR pair
    if (rows == TB) {
        // Whole tile contiguous: TB*NV*4 = 14976 bytes = 936 x B128 copies.
        const int nchunks = (TB * NV) / 4;           // 936 float4 chunks
        for (int i = tid; i < nchunks; i += THREADS) {
            unsigned lds_off = (unsigned)(size_t)(&s_rows[i * 4]);
            unsigned goff    = (unsigned)(i * 16);
            asm volatile("global_load_async_to_lds_b128 %0, %1, %2"
                         :: "v"(lds_off), "v"(goff), "s"(gbase) : "memory");
        }
    } else {
        // Generic tail tile: one float per async op (not hit at BATCH=131072).
        const int nfloats = rows * NV;
        for (int i = tid; i < nfloats; i += THREADS) {
            unsigned lds_off = (unsigned)(size_t)(&s_rows[i]);
            unsigned goff    = (unsigned)(i * 4);
            asm volatile("global_load_async_to_lds_b32 %0, %1, %2"
                         :: "v"(lds_off), "v"(goff), "s"(gbase) : "memory");
        }
    }

    // One wait covers index map + row tile; barrier publishes LDS to all waves.
    asm volatile("s_wait_asynccnt 0" ::: "memory");
    __syncthreads();

    // ---- gather from LDS, write coalesced NT float4 stores ----
    // Per block: rows * 150 float4 stores; consecutive lanes -> consecutive
    // 16B addresses (one wave32 streams 512 contiguous bytes per store op).
    const int c4_per_row = NH / 4;                   // 150
    const int chunks     = rows * c4_per_row;
    for (int i = tid; i < chunks; i += THREADS) {
        const int r  = i / c4_per_row;
        const int c4 = i - r * c4_per_row;
        const int h  = c4 * 4;
        const float* rowp = &s_rows[r * NV];
        v4f v;
        v.x = rowp[s_idx[h + 0]];
        v.y = rowp[s_idx[h + 1]];
        v.z = rowp[s_idx[h + 2]];
        v.w = rowp[s_idx[h + 3]];
        v4f* dst = (v4f*)(out + (size_t)(row0 + r) * NH + h);
        __builtin_nontemporal_store(v, dst);         // stream past L2
    }
}

extern "C" void kernel_launch(void* const* d_in, const int* in_sizes, int n_in,
                              void* d_out, int out_size, void* d_ws, size_t ws_size,
                              hipStream_t stream)
{
    const float* verb = (const float*)d_in[0];   // (BATCH, 117) f32
    const int*   map  = (const int*)d_in[1];     // (600,) int
    float*       out  = (float*)d_out;           // (BATCH, 600) f32

    const int batch  = in_sizes[0] / NV;         // 131072
    const int blocks = (batch + TB - 1) / TB;    // 4096

    ScatterVerbsToHois_kernel<<<blocks, THREADS, 0, stream>>>(verb, map, out, batch);
}